// RecSysGNN_47467978556189
// MI455X (gfx1250) — compile-verified
//
#include <hip/hip_runtime.h>
#include <hip/hip_bf16.h>

// ---------------------------------------------------------------------------
// RecSysGNN propagation for gfx1250 (MI455X).
//
// Key identity: leaky_relu(x, 0.01) = 0.505*x + 0.495*|x|, and
// |a*w| = |a|*|w|, so the per-edge/per-feature attention scores are an EXACT
// rank-4 GEMM:  S[16 edges x 64 feats] = P[16x4] * W[4x64]
// -> V_WMMA_F32_16X16X4_F32, 4 feature tiles per 16-edge group.
//
// Per layer (2 edge passes + cheap node passes):
//   1) scores via WMMA -> atomic max into m[dst,f]   (int-encoded f32 max)
//   2) decode m (empty segment -> 0, matches jnp.where(isfinite(m),m,0))
//   3) scores via WMMA -> w = exp(s - m[dst,f]);
//      atomicAdd den[dst,f] += w;  atomicAdd num[dst,f] += norm_e*w*x[src,f]
//   4) x_next = num/(den+eps)  (softmax denominator applied at node level),
//      out += x_next
// Node arrays (~150MB hot) stay resident in the 192MB L2; atomics hit L2.
// ---------------------------------------------------------------------------

typedef __attribute__((ext_vector_type(2))) float v2f;
typedef __attribute__((ext_vector_type(8))) float v8f;

#define LAT 64
#define LRELU_A 0.505f   // (1+slope)/2, slope=0.01
#define LRELU_B 0.495f   // (1-slope)/2
#define GNN_EPS 1e-16f
#define PF_DIST 4096

#if __has_builtin(__builtin_amdgcn_wmma_f32_16x16x4_f32)
#define HAVE_WMMA4 1
#else
#define HAVE_WMMA4 0
#endif

__device__ __forceinline__ float lrelu_f(float x) {
    return LRELU_A * x + LRELU_B * fabsf(x);
}
// Order-preserving int encoding of f32 (for atomicMax on signed int).
__device__ __forceinline__ int enc_f32(float s) {
    int i = __float_as_int(s);
    return (i >= 0) ? i : (i ^ 0x7fffffff);
}
__device__ __forceinline__ float dec_f32(int k) {
    return __int_as_float((k >= 0) ? k : (k ^ 0x7fffffff));
}

// ---- init: out = emb, deg = 0 ------------------------------------------------
__global__ void k_init(const float* __restrict__ emb, float* __restrict__ out,
                       unsigned* __restrict__ deg, int N) {
    int i = blockIdx.x * blockDim.x + threadIdx.x;
    int total = N * LAT;
    if (i < total) out[i] = emb[i];
    if (i < N) deg[i] = 0u;
}

// ---- degree ------------------------------------------------------------------
__global__ void k_deg(const int* __restrict__ dst, unsigned* __restrict__ deg, int E) {
    int e = blockIdx.x * blockDim.x + threadIdx.x;
    if (e < E) atomicAdd(&deg[dst[e]], 1u);
}

// ---- dinv = deg > 0 ? deg^-0.5 : 0   (in place: uint -> float) --------------
__global__ void k_dinv(float* __restrict__ dinv, int N) {
    int i = blockIdx.x * blockDim.x + threadIdx.x;
    if (i < N) {
        unsigned d = ((const unsigned*)dinv)[i];
        dinv[i] = d ? (1.0f / sqrtf((float)d)) : 0.0f;
    }
}

// ---- per-edge symmetric norm -------------------------------------------------
__global__ void k_norm(const int* __restrict__ src, const int* __restrict__ dst,
                       const float* __restrict__ dinv, float* __restrict__ normE, int E) {
    int e = blockIdx.x * blockDim.x + threadIdx.x;
    if (e < E) normE[e] = dinv[src[e]] * dinv[dst[e]];
}

// ---- per-layer clear ---------------------------------------------------------
__global__ void k_clear(int* __restrict__ mkeys, float* __restrict__ den,
                        float* __restrict__ num, int total) {
    int i = blockIdx.x * blockDim.x + threadIdx.x;
    if (i < total) {
        mkeys[i] = (int)0x80000000;  // encodes "empty segment"
        den[i] = 0.0f;
        num[i] = 0.0f;
    }
}

// ---- pass 1: WMMA scores -> segment max --------------------------------------
__global__ void k_scoremax(const float* __restrict__ attrs, const int* __restrict__ dst,
                           const float* __restrict__ aw, const float* __restrict__ rw,
                           int* __restrict__ mkeys, int E) {
    int gtid = blockIdx.x * blockDim.x + threadIdx.x;
    int wave = gtid >> 5;
    int lane = threadIdx.x & 31;
    int ebase = wave * 16;
    if (ebase >= E) return;
    int m16 = lane & 15;
    int hseg = lane >> 4;  // 0: K=0,1 (a-attr) half, 1: K=2,3 (r-attr) half
#if HAVE_WMMA4
    if (ebase + 16 <= E) {
        int e = ebase + m16;
        float pa = attrs[2 * e];
        float pr = attrs[2 * e + 1];
        float p = hseg ? pr : pa;
        v2f a;                       // A 16x4: lanes0-15 {K0,K1}, lanes16-31 {K2,K3}
        a.x = LRELU_A * p;
        a.y = LRELU_B * fabsf(p);
        int dste = dst[e];
        int dd[8];
#pragma unroll
        for (int i = 0; i < 8; ++i)
            dd[i] = __shfl(dste, i + hseg * 8, 32) * LAT;
#pragma unroll
        for (int t = 0; t < 4; ++t) {
            int f = t * 16 + m16;
            float wv = hseg ? rw[f] : aw[f];
            v2f b;                   // B 4x16: lanes0-15 {K0,K1}, lanes16-31 {K2,K3}
            b.x = wv;
            b.y = fabsf(wv);
            v8f c = {};
            v8f d = __builtin_amdgcn_wmma_f32_16x16x4_f32(
                false, a, false, b, (short)0, c, false, false);
#pragma unroll
            for (int i = 0; i < 8; ++i)  // D vgpr i: edge (i + 8*hseg), feature f
                atomicMax(&mkeys[dd[i] + f], enc_f32(d[i]));
        }
        return;
    }
#endif
    // scalar tail (also full fallback if builtin missing)
    int cnt = E - ebase;
    if (cnt > 16) cnt = 16;
    for (int idx = lane; idx < cnt * LAT; idx += 32) {
        int el = idx >> 6, f = idx & 63;
        int e = ebase + el;
        float pa = attrs[2 * e], pr = attrs[2 * e + 1];
        float s = lrelu_f(pa * aw[f]) + lrelu_f(pr * rw[f]);
        atomicMax(&mkeys[dst[e] * LAT + f], enc_f32(s));
    }
}

// ---- decode max keys -> float (empty segment -> 0) ---------------------------
__global__ void k_decode(float* __restrict__ mf, int total) {
    int i = blockIdx.x * blockDim.x + threadIdx.x;
    if (i < total) {
        int k = ((const int*)mf)[i];
        mf[i] = (k == (int)0x80000000) ? 0.0f : dec_f32(k);
    }
}

// ---- pass 2: WMMA scores -> exp, den & weighted-message scatter --------------
__global__ void k_main(const float* __restrict__ attrs, const int* __restrict__ src,
                       const int* __restrict__ dst,
                       const float* __restrict__ aw, const float* __restrict__ rw,
                       const float* __restrict__ mflt, const float* __restrict__ normE,
                       const float* __restrict__ xcur,
                       float* __restrict__ den, float* __restrict__ num, int E) {
    int gtid = blockIdx.x * blockDim.x + threadIdx.x;
    int wave = gtid >> 5;
    int lane = threadIdx.x & 31;
    int ebase = wave * 16;
    if (ebase >= E) return;
    int m16 = lane & 15;
    int hseg = lane >> 4;
#if HAVE_WMMA4
    if (ebase + 16 <= E) {
        int e = ebase + m16;
        float pa = attrs[2 * e];
        float pr = attrs[2 * e + 1];
        float p = hseg ? pr : pa;
        v2f a;
        a.x = LRELU_A * p;
        a.y = LRELU_B * fabsf(p);
        int dste = dst[e];
        int srce = src[e];
        float nrme = normE[e];
        // Pre-warm a future gather row (global_prefetch_b8).
        int pe = e + PF_DIST;
        if (pe < E) {
            int ps = src[pe];
            __builtin_prefetch(&xcur[ps * LAT], 0, 1);
        }
        int dd[8], ss[8];
        float nn[8];
#pragma unroll
        for (int i = 0; i < 8; ++i) {
            int el = i + hseg * 8;
            dd[i] = __shfl(dste, el, 32) * LAT;
            ss[i] = __shfl(srce, el, 32) * LAT;
            nn[i] = __shfl(nrme, el, 32);
        }
#pragma unroll
        for (int t = 0; t < 4; ++t) {
            int f = t * 16 + m16;
            float wv = hseg ? rw[f] : aw[f];
            v2f b;
            b.x = wv;
            b.y = fabsf(wv);
            v8f c = {};
            v8f d = __builtin_amdgcn_wmma_f32_16x16x4_f32(
                false, a, false, b, (short)0, c, false, false);
#pragma unroll
            for (int i = 0; i < 8; ++i) {
                float w = __expf(d[i] - mflt[dd[i] + f]);
                atomicAdd(&den[dd[i] + f], w);
                atomicAdd(&num[dd[i] + f], nn[i] * w * xcur[ss[i] + f]);
            }
        }
        return;
    }
#endif
    int cnt = E - ebase;
    if (cnt > 16) cnt = 16;
    for (int idx = lane; idx < cnt * LAT; idx += 32) {
        int el = idx >> 6, f = idx & 63;
        int e = ebase + el;
        float pa = attrs[2 * e], pr = attrs[2 * e + 1];
        float s = lrelu_f(pa * aw[f]) + lrelu_f(pr * rw[f]);
        int d64 = dst[e] * LAT;
        float w = __expf(s - mflt[d64 + f]);
        atomicAdd(&den[d64 + f], w);
        atomicAdd(&num[d64 + f], normE[e] * w * xcur[src[e] * LAT + f]);
    }
}

// ---- node finalize: x = num/(den+eps); out += x; num becomes next x ----------
__global__ void k_finalize(float* __restrict__ num, const float* __restrict__ den,
                           float* __restrict__ out, int total) {
    int i = blockIdx.x * blockDim.x + threadIdx.x;
    if (i < total) {
        float x = num[i] / (den[i] + GNN_EPS);
        num[i] = x;
        out[i] += x;
    }
}

__global__ void k_scale(float* __restrict__ out, float s, int total) {
    int i = blockIdx.x * blockDim.x + threadIdx.x;
    if (i < total) out[i] *= s;
}

// ---------------------------------------------------------------------------
extern "C" void kernel_launch(void* const* d_in, const int* in_sizes, int n_in,
                              void* d_out, int out_size, void* d_ws, size_t ws_size,
                              hipStream_t stream) {
    const int* edge_index = (const int*)d_in[0];   // [2, E] int32
    const float* edge_attrs = (const float*)d_in[1];  // [E, 2]
    const float* emb = (const float*)d_in[2];      // [N, 64]
    const float* a_att = (const float*)d_in[3];    // [L, 64]
    const float* r_att = (const float*)d_in[4];    // [L, 64]

    const int E = in_sizes[0] / 2;
    const int N = in_sizes[2] / LAT;
    const int NL = in_sizes[3] / LAT;
    const int* src = edge_index;
    const int* dst = edge_index + E;
    float* out = (float*)d_out;

    // Workspace carve-up (~208 MB for N=200K, E=500K).
    char* ws = (char*)d_ws;
    const size_t nodeF = (size_t)N * LAT * sizeof(float);
    float* bufA = (float*)ws;  ws += nodeF;
    float* bufB = (float*)ws;  ws += nodeF;
    float* mbuf = (float*)ws;  ws += nodeF;   // int max-keys, then decoded float m
    float* den  = (float*)ws;  ws += nodeF;
    float* dinv = (float*)ws;  ws += (size_t)N * sizeof(float);  // aliased uint deg
    float* normE = (float*)ws; ws += (size_t)E * sizeof(float);

    const int total = N * LAT;
    const int gTotal = (total + 255) / 256;
    const int gN = (N + 255) / 256;
    const int gE = (E + 255) / 256;
    const long ewaves = (long)(E + 15) / 16;          // one wave per 16 edges
    const int gEW = (int)((ewaves * 32 + 255) / 256);

    k_init<<<gTotal, 256, 0, stream>>>(emb, out, (unsigned*)dinv, N);
    k_deg<<<gE, 256, 0, stream>>>(dst, (unsigned*)dinv, E);
    k_dinv<<<gN, 256, 0, stream>>>(dinv, N);
    k_norm<<<gE, 256, 0, stream>>>(src, dst, dinv, normE, E);

    const float* xcur = emb;
    for (int l = 0; l < NL; ++l) {
        float* num = (l & 1) ? bufB : bufA;
        k_clear<<<gTotal, 256, 0, stream>>>((int*)mbuf, den, num, total);
        k_scoremax<<<gEW, 256, 0, stream>>>(edge_attrs, dst, a_att + l * LAT,
                                            r_att + l * LAT, (int*)mbuf, E);
        k_decode<<<gTotal, 256, 0, stream>>>(mbuf, total);
        k_main<<<gEW, 256, 0, stream>>>(edge_attrs, src, dst, a_att + l * LAT,
                                        r_att + l * LAT, mbuf, normE, xcur, den, num, E);
        k_finalize<<<gTotal, 256, 0, stream>>>(num, den, out, total);
        xcur = num;
    }
    k_scale<<<gTotal, 256, 0, stream>>>(out, 1.0f / (float)(NL + 1), total);
}